// SimpleRPMGLoss_74706661146954
// MI455X (gfx1250) — compile-verified
//
#include <hip/hip_runtime.h>
#include <math.h>

// ---------------------------------------------------------------------------
// SimpleRPMGLoss for MI455X (gfx1250, wave32).
// Memory-bound reduction: 100.7 MB read -> 1 float (~4.3 us at 23.3 TB/s).
//   * float4 (global_load_b128) loads, 2 rows (48B, 16B-aligned) per thread/iter
//   * hardware v_sin_f32 / v_cos_f32 via __sinf/__cosf (TRANS, co-execute)
//   * wave32 cross-lane reduction with one V_WMMA_F32_16X16X4_F32 + one
//     xor-16 bpermute; no broadcast shuffles (lane0 holds trans total,
//     lane1 holds rot total directly).
// ---------------------------------------------------------------------------

typedef __attribute__((ext_vector_type(2))) float v2f;
typedef __attribute__((ext_vector_type(8))) float v8f;

#define ANGLE_WEIGHT_C 100.0f
#define EPS_C 1e-07f

// Per-row contribution: smooth-L1 over 3 translation comps into tacc,
// geodesic angle between Euler-XYZ rotations into racc.
__device__ __forceinline__ void row_contrib(
    float p0, float p1, float p2, float p3, float p4, float p5,
    float t0, float t1, float t2, float t3, float t4, float t5,
    float& tacc, float& racc)
{
  // smooth L1 (beta = 1)
  {
    float d0 = p0 - t0, d1 = p1 - t1, d2 = p2 - t2;
    float a0 = fabsf(d0), a1 = fabsf(d1), a2 = fabsf(d2);
    tacc += (a0 < 1.0f) ? 0.5f * d0 * d0 : a0 - 0.5f;
    tacc += (a1 < 1.0f) ? 0.5f * d1 * d1 : a1 - 0.5f;
    tacc += (a2 < 1.0f) ? 0.5f * d2 * d2 : a2 - 0.5f;
  }

  // hardware trig: v_sin_f32 / v_cos_f32 (inputs ~N(0,1), well in range)
  float ca = __cosf(p3), sa = __sinf(p3);
  float cb = __cosf(p4), sb = __sinf(p4);
  float cc = __cosf(p5), sc = __sinf(p5);
  float ga = __cosf(t3), ha = __sinf(t3);
  float gb = __cosf(t4), hb = __sinf(t4);
  float gc = __cosf(t5), hc = __sinf(t5);

  // predicted rotation matrix (Euler XYZ)
  float r00 = cb * cc;
  float r01 = cc * sa * sb - ca * sc;
  float r02 = sc * sa + ca * cc * sb;
  float r10 = cb * sc;
  float r11 = ca * cc + sa * sb * sc;
  float r12 = ca * sb * sc - cc * sa;
  float r20 = -sb;
  float r21 = cb * sa;
  float r22 = ca * cb;
  // target rotation matrix
  float q00 = gb * gc;
  float q01 = gc * ha * hb - ga * hc;
  float q02 = hc * ha + ga * gc * hb;
  float q10 = gb * hc;
  float q11 = ga * gc + ha * hb * hc;
  float q12 = ga * hb * hc - gc * ha;
  float q20 = -hb;
  float q21 = gb * ha;
  float q22 = ga * gb;

  float trace = r00 * q00 + r01 * q01 + r02 * q02
              + r10 * q10 + r11 * q11 + r12 * q12
              + r20 * q20 + r21 * q21 + r22 * q22;

  float cosang = 0.5f * (trace - 1.0f);
  cosang = fminf(fmaxf(cosang, -1.0f + EPS_C), 1.0f - EPS_C);
  racc += acosf(cosang);
}

// Reduce (tacc, racc) across the 32 lanes of a wave with one
// V_WMMA_F32_16X16X4_F32 plus a single xor-16 lane swap.
//
// A layout (f32 16x4): lanes 0-15 hold A[lane][K] with VGPR0=K0, VGPR1=K1;
// lanes 16-31 hold A[lane-16][K] with VGPR0=K2, VGPR1=K3.
//   -> per-lane A = {tacc, racc}: trans lands in K{0,2}, rot in K{1,3}.
// B layout (f32 4x16): lanes 0-15 N=lane (VGPR0=K0 row, VGPR1=K1 row),
// lanes 16-31 N=lane-16 (VGPR0=K2 row, VGPR1=K3 row).
//   -> b.x = ((lane&15)==0) sets B[0][0]=B[2][0]=1 ; b.y likewise for col 1.
// So D[m][0] = trans_m + trans_{m+16}, D[m][1] = rot_m + rot_{m+16}.
// D layout: lane n (0-15) VGPR r = D[r][n]; lane n+16 VGPR r = D[r+8][n].
// Per-lane sum of the 8 D regs + xor-16 add => lane (n, n+16) both hold
// colsum[n]: column 0 = wave trans total, column 1 = wave rot total.
// NO broadcast: caller uses lane 0 (trans) and lane 1 (rot) directly.
__device__ __forceinline__ float wave_reduce_col(float tacc, float racc)
{
  unsigned lane = threadIdx.x & 31u;
  v2f a;
  a.x = tacc;
  a.y = racc;
  v2f b;
  b.x = ((lane & 15u) == 0u) ? 1.0f : 0.0f;
  b.y = ((lane & 15u) == 1u) ? 1.0f : 0.0f;
  v8f c = {};
  c = __builtin_amdgcn_wmma_f32_16x16x4_f32(
      /*neg_a=*/false, a, /*neg_b=*/false, b,
      /*c_mod=*/(short)0, c, /*reuse_a=*/false, /*reuse_b=*/false);
  float s = c[0] + c[1] + c[2] + c[3] + c[4] + c[5] + c[6] + c[7];
  return s + __shfl_xor(s, 16, 32);   // colsum for column (lane&15)
}

__global__ void __launch_bounds__(256)
rpmg_partial_kernel(const float* __restrict__ pred,
                    const float* __restrict__ tgt,
                    float* __restrict__ part, long rows)
{
  const long pairs  = rows >> 1;
  const long stride = (long)gridDim.x * (long)blockDim.x;
  const long tid    = (long)blockIdx.x * (long)blockDim.x + threadIdx.x;

  float tacc = 0.0f, racc = 0.0f;
  const float4* __restrict__ p4 = (const float4*)pred;
  const float4* __restrict__ t4 = (const float4*)tgt;

  for (long i = tid; i < pairs; i += stride) {
    float4 a0 = p4[i * 3 + 0];
    float4 a1 = p4[i * 3 + 1];
    float4 a2 = p4[i * 3 + 2];
    float4 b0 = t4[i * 3 + 0];
    float4 b1 = t4[i * 3 + 1];
    float4 b2 = t4[i * 3 + 2];
    row_contrib(a0.x, a0.y, a0.z, a0.w, a1.x, a1.y,
                b0.x, b0.y, b0.z, b0.w, b1.x, b1.y, tacc, racc);
    row_contrib(a1.z, a1.w, a2.x, a2.y, a2.z, a2.w,
                b1.z, b1.w, b2.x, b2.y, b2.z, b2.w, tacc, racc);
  }

  // odd tail row (not hit for B = 2M, kept for generality)
  if ((rows & 1) && tid == 0) {
    const float* pp = pred + (rows - 1) * 6;
    const float* tt = tgt + (rows - 1) * 6;
    row_contrib(pp[0], pp[1], pp[2], pp[3], pp[4], pp[5],
                tt[0], tt[1], tt[2], tt[3], tt[4], tt[5], tacc, racc);
  }

  float col = wave_reduce_col(tacc, racc);   // EXEC all-ones (reconverged)

  __shared__ float lt[8];
  __shared__ float lr[8];
  const unsigned lane = threadIdx.x & 31u;
  const unsigned wave = threadIdx.x >> 5;
  if (lane == 0u)      lt[wave] = col;       // wave trans total
  else if (lane == 1u) lr[wave] = col;       // wave rot total
  __syncthreads();
  if (threadIdx.x == 0) {
    float ts = 0.0f, rs = 0.0f;
    const int nw = blockDim.x >> 5;
    for (int k = 0; k < nw; ++k) { ts += lt[k]; rs += lr[k]; }
    part[2 * blockIdx.x + 0] = ts;   // overwrite: no ws init required
    part[2 * blockIdx.x + 1] = rs;
  }
}

__global__ void __launch_bounds__(256)
rpmg_final_kernel(const float* __restrict__ part, int nparts,
                  float* __restrict__ out, float rows)
{
  float tacc = 0.0f, racc = 0.0f;
  for (int i = threadIdx.x; i < nparts; i += (int)blockDim.x) {
    tacc += part[2 * i + 0];
    racc += part[2 * i + 1];
  }

  float col = wave_reduce_col(tacc, racc);

  __shared__ float lt[8];
  __shared__ float lr[8];
  const unsigned lane = threadIdx.x & 31u;
  const unsigned wave = threadIdx.x >> 5;
  if (lane == 0u)      lt[wave] = col;
  else if (lane == 1u) lr[wave] = col;
  __syncthreads();
  if (threadIdx.x == 0) {
    float ts = 0.0f, rs = 0.0f;
    const int nw = blockDim.x >> 5;
    for (int k = 0; k < nw; ++k) { ts += lt[k]; rs += lr[k]; }
    out[0] = ts / (3.0f * rows) + ANGLE_WEIGHT_C * (rs / rows);
  }
}

extern "C" void kernel_launch(void* const* d_in, const int* in_sizes, int n_in,
                              void* d_out, int out_size, void* d_ws, size_t ws_size,
                              hipStream_t stream) {
  const float* pred = (const float*)d_in[0];
  const float* tgt  = (const float*)d_in[1];
  float* out = (float*)d_out;
  float* part = (float*)d_ws;

  const long rows = (long)in_sizes[0] / 6;   // B = 2,097,152

  int nblocks = 2048;                        // 524288 threads -> 2 pairs each
  const size_t need = (size_t)nblocks * 2 * sizeof(float);
  if (need > ws_size) {
    size_t fit = ws_size / (2 * sizeof(float));
    nblocks = (fit < 1) ? 1 : (int)((fit > 2048) ? 2048 : fit);
  }

  rpmg_partial_kernel<<<nblocks, 256, 0, stream>>>(pred, tgt, part, rows);
  rpmg_final_kernel<<<1, 256, 0, stream>>>(part, nblocks, out, (float)rows);
}